// PMLoss_9500467658994
// MI455X (gfx1250) — compile-verified
//
#include <hip/hip_runtime.h>

// ---------------------------------------------------------------------------
// PMLoss on gfx1250 (MI455X).
//
// Math: xp - xt == (R_pred - R_tgt) @ p + (t_pred - t_tgt): the per-point
// work collapses to one affine transform d = M'*p + dt' with the smooth-L1
// weight w folded in.  Stage 1 (one thread per batch item) does the
// quaternion algebra and emits a 4x16 f32 WMMA B-fragment per batch item:
//     B[k][n] = M'[n][k] (k<3), dt'[n] (k==3)  for n in 0..2, zero for n>=3
// in wave32 B-operand layout (VGPR0: K0 lanes0-15 / K2 lanes16-31;
// VGPR1: K1 / K3).  Stage 2 runs V_WMMA_F32_16X16X4_F32 per 16-point tile
// with A rows = (x,y,z,1); D columns 3..15 are exactly zero and
// smoothL1(0)==0, so the elementwise loss + reduction covers the full 16x16
// tile with no predication.
//
// The wave's tile index is pinned to an SGPR via readfirstlane so the tile
// loop is a *scalar* loop: EXEC is structurally all-ones around every WMMA
// (ISA requirement) and loop bookkeeping is SALU-only.
//
// Roofline: points = 22*65536*3*4B = 17 MB -> resident in the 192 MB L2; the
// 128 * 0.75 MB logical reads are L2 hits.  ~0.5 GFLOP total -> the kernel is
// latency-bound; exact-f32 WMMA matches the f32 reference numerics for free.
// ---------------------------------------------------------------------------

typedef __attribute__((ext_vector_type(2))) float v2f;
typedef __attribute__((ext_vector_type(8))) float v8f;

// ---------------- Stage 1: per-batch pose algebra -> B-fragments ------------
__global__ void pmloss_stage1(const float* __restrict__ rotations,     // (B, 4C)
                              const float* __restrict__ translations,  // (B, 3C)
                              const float* __restrict__ poses_src,     // (B, 9)
                              const float* __restrict__ poses_tgt,     // (B, 9)
                              const float* __restrict__ extents,       // (C, 3)
                              float* __restrict__ ws,                  // B * 64 floats
                              float* __restrict__ out,                 // scalar
                              int B, int C) {
  int b = blockIdx.x * blockDim.x + threadIdx.x;
  if (b == 0) out[0] = 0.0f;  // stream-ordered before stage 2's atomics
  if (b >= B) return;

  const float* ps = poses_src + b * 9;
  const float* pt = poses_tgt + b * 9;
  int cls = (int)ps[1];

  // q_delta = normalize(rotations[b, cls])
  const float* qr = rotations + (size_t)b * 4 * C + 4 * cls;
  float dw = qr[0], dx = qr[1], dy = qr[2], dz = qr[3];
  float inv = 1.0f / sqrtf(dw * dw + dx * dx + dy * dy + dz * dz);
  dw *= inv; dx *= inv; dy *= inv; dz *= inv;

  // q_pred = q_delta * q_src
  float sw = ps[2], sx = ps[3], sy = ps[4], sz = ps[5];
  float pw = dw * sw - dx * sx - dy * sy - dz * sz;
  float px = dw * sx + dx * sw + dy * sz - dz * sy;
  float py = dw * sy - dx * sz + dy * sw + dz * sx;
  float pz = dw * sz + dx * sy - dy * sx + dz * sw;

  // t_pred from translation head
  const float* v = translations + (size_t)b * 3 * C + 3 * cls;
  float ts0 = ps[6], ts1 = ps[7], ts2 = ps[8];
  float tz = ts2 / expf(v[2]);
  float tx = (v[0] + ts0 / ts2) * tz;
  float ty = (v[1] + ts1 / ts2) * tz;

  // R_pred, R_tgt (quat -> 3x3)
  float Rp[3][3], Rt[3][3];
  {
    float w = pw, x = px, y = py, z = pz;
    Rp[0][0] = 1.f - 2.f * (y * y + z * z); Rp[0][1] = 2.f * (x * y - w * z); Rp[0][2] = 2.f * (x * z + w * y);
    Rp[1][0] = 2.f * (x * y + w * z); Rp[1][1] = 1.f - 2.f * (x * x + z * z); Rp[1][2] = 2.f * (y * z - w * x);
    Rp[2][0] = 2.f * (x * z - w * y); Rp[2][1] = 2.f * (y * z + w * x); Rp[2][2] = 1.f - 2.f * (x * x + y * y);
  }
  {
    float w = pt[2], x = pt[3], y = pt[4], z = pt[5];
    Rt[0][0] = 1.f - 2.f * (y * y + z * z); Rt[0][1] = 2.f * (x * y - w * z); Rt[0][2] = 2.f * (x * z + w * y);
    Rt[1][0] = 2.f * (x * y + w * z); Rt[1][1] = 1.f - 2.f * (x * x + z * z); Rt[1][2] = 2.f * (y * z - w * x);
    Rt[2][0] = 2.f * (x * z - w * y); Rt[2][1] = 2.f * (y * z + w * x); Rt[2][2] = 1.f - 2.f * (x * x + y * y);
  }

  // w = 10 / ||extents[cls]||, folded into M' and dt'
  const float* ex = extents + 3 * cls;
  float wgt = 10.0f / sqrtf(ex[0] * ex[0] + ex[1] * ex[1] + ex[2] * ex[2]);

  float Mp[3][3], dtp[3];
  for (int i = 0; i < 3; ++i)
    for (int j = 0; j < 3; ++j)
      Mp[i][j] = wgt * (Rp[i][j] - Rt[i][j]);
  float tp[3] = {tx, ty, tz};
  for (int i = 0; i < 3; ++i) dtp[i] = wgt * (tp[i] - pt[6 + i]);

  // Emit wave32 B-operand fragment (4x16 f32): 2 VGPRs x 32 lanes.
  float* w64 = ws + (size_t)b * 64;
  for (int lane = 0; lane < 32; ++lane) {
    int n = lane & 15;
    int kbase = (lane >> 4) * 2;  // 0 or 2
    float b0 = 0.0f, b1 = 0.0f;
    if (n < 3) {
      b0 = (kbase + 0 < 3) ? Mp[n][kbase + 0] : dtp[n];
      b1 = (kbase + 1 < 3) ? Mp[n][kbase + 1] : dtp[n];
    }
    w64[lane] = b0;
    w64[32 + lane] = b1;
  }
}

// ---------------- Stage 2: WMMA point transform + smooth-L1 reduce ----------
__device__ __forceinline__ float tile_loss(const float* __restrict__ pts,
                                           int p0, int lane, v2f bf,
                                           const v8f& czero) {
  const int m = lane & 15;
  const size_t base = (size_t)(p0 + m) * 3;

  // A fragment (16x4 f32): lanes 0-15 carry (x,y) = K0,K1; lanes 16-31 carry
  // (z, 1.0) = K2,K3 of the same 16 rows.
  v2f af;
  if (lane < 16) {
    af.x = pts[base + 0];
    af.y = pts[base + 1];
  } else {
    af.x = pts[base + 2];
    af.y = 1.0f;
  }

  // D = A * B  (exact f32 path: v_wmma_f32_16x16x4_f32)
  v8f d = __builtin_amdgcn_wmma_f32_16x16x4_f32(
      false, af, false, bf, (short)0, czero, false, false);

  // Smooth-L1 over the whole tile; padded columns are exactly 0 -> loss 0.
  float acc = 0.0f;
#pragma unroll
  for (int r = 0; r < 8; ++r) {
    float x = d[r];
    float ax = fabsf(x);
    acc += (ax < 1.0f) ? 0.5f * x * x : (ax - 0.5f);
  }
  return acc;
}

__global__ void __launch_bounds__(256)
pmloss_stage2(const float* __restrict__ points,     // (C, P, 3)
              const float* __restrict__ poses_src,  // (B, 9) -> cls
              const float* __restrict__ ws,         // B-fragments
              float* __restrict__ out,
              int P, int numTiles, int tilesPerBlock, float scale) {
  const int b = blockIdx.x;
  const int lane = threadIdx.x & 31;
  // Pin the wave index to an SGPR: the tile loop below becomes a scalar loop
  // (SALU branch, EXEC untouched -> all-ones at every WMMA, per ISA rules).
  const int wv = __builtin_amdgcn_readfirstlane((int)(threadIdx.x >> 5));

  const int cls = (int)poses_src[b * 9 + 1];
  const float* pts = points + (size_t)cls * (size_t)P * 3;

  // Per-batch B fragment (uniform across the batch's waves).
  v2f bf;
  bf.x = ws[(size_t)b * 64 + lane];
  bf.y = ws[(size_t)b * 64 + 32 + lane];

  const int tBeg = blockIdx.y * tilesPerBlock;
  int tEnd = tBeg + tilesPerBlock;
  if (tEnd > numTiles) tEnd = numTiles;

  // Scalar trip count for this wave (tiles strided by 8 waves).
  int cnt = tEnd - (tBeg + wv);
  cnt = (cnt > 0) ? ((cnt + 7) >> 3) : 0;

  const v8f czero = {};
  float acc = 0.0f;

  int t = tBeg + wv;
  // Pairs of tiles: two independent WMMAs in flight, each tile's loss VALU
  // covers the other's WMMA->VALU hazard window.
  for (int i = 0; i + 1 < cnt; i += 2, t += 16) {
    // Prefetch the next pair's first tile into the near caches (L2/WGP path).
    __builtin_prefetch(&pts[(size_t)(t + 16) * 16 * 3], 0, 3);
    acc += tile_loss(pts, t * 16, lane, bf, czero);
    acc += tile_loss(pts, (t + 8) * 16, lane, bf, czero);
  }
  if (cnt & 1) {
    acc += tile_loss(pts, (tBeg + wv + (cnt - 1) * 8) * 16, lane, bf, czero);
  }

  // Wave reduction (wave32!), then block reduction, one atomic per block.
  for (int off = 16; off > 0; off >>= 1) acc += __shfl_xor(acc, off, 32);

  __shared__ float wsum[8];
  if (lane == 0) wsum[wv] = acc;
  __syncthreads();
  if (threadIdx.x == 0) {
    float s = 0.0f;
#pragma unroll
    for (int i = 0; i < 8; ++i) s += wsum[i];
    atomicAdd(out, s * scale);
  }
}

// ---------------------------------------------------------------------------
extern "C" void kernel_launch(void* const* d_in, const int* in_sizes, int n_in,
                              void* d_out, int out_size, void* d_ws, size_t ws_size,
                              hipStream_t stream) {
  const float* rotations    = (const float*)d_in[0];
  const float* translations = (const float*)d_in[1];
  const float* poses_src    = (const float*)d_in[2];
  const float* poses_tgt    = (const float*)d_in[3];
  const float* extents      = (const float*)d_in[4];
  const float* points       = (const float*)d_in[5];
  float* out = (float*)d_out;
  float* ws  = (float*)d_ws;   // needs B*64 floats (32 KB for B=128)

  const int B = in_sizes[2] / 9;                       // poses_src: (B, 9)
  const int C = in_sizes[4] / 3;                       // extents:   (C, 3)
  const long long P = (long long)in_sizes[5] / (3LL * C);  // points: (C, P, 3)

  const float scale = 1.0f / ((float)B * (float)P);

  pmloss_stage1<<<dim3((B + 127) / 128), 128, 0, stream>>>(
      rotations, translations, poses_src, poses_tgt, extents, ws, out, B, C);

  const int numTiles = (int)(P / 16);                  // P = 65536 -> 4096
  const int SPLIT = 32;                                // blocks per batch item
  const int tilesPerBlock = (numTiles + SPLIT - 1) / SPLIT;

  pmloss_stage2<<<dim3(B, SPLIT), 256, 0, stream>>>(
      points, poses_src, ws, out, (int)P, numTiles, tilesPerBlock, scale);
}